// PCEN_84791244357836
// MI455X (gfx1250) — compile-verified
//
#include <hip/hip_runtime.h>

typedef float v2f __attribute__((ext_vector_type(2)));
typedef float v8f __attribute__((ext_vector_type(8)));

#define T_LEN   6000
#define TILE    256      // 16 segments x 16 positions per wave-tile
#define NT      ((T_LEN + TILE - 1) / TILE)   // 24 (last tile partial: 112 elems)

// integer power of a (a in (0,1)); e<0 -> 0 (strictly-lower-triangular zero fill)
__device__ __forceinline__ float apow_int(float a, int e) {
    if (e < 0) return 0.0f;
    float p = 1.0f;
    for (int i = 0; i < e; ++i) p *= a;
    return p;
}

__device__ __forceinline__ float lane_read(float v, int srcLane) {
    return __uint_as_float(__builtin_amdgcn_readlane(__float_as_uint(v), srcLane));
}

// One 256-element tile of the blocked linear scan + pointwise PCEN.
// Returns the carry (smoother value after the tile's last element).
template <bool TAIL>
__device__ __forceinline__ float pcen_tile(
    const float* __restrict__ xrow, float* __restrict__ orow,
    int t0, float carry, const v2f* wb,
    float s, float q, float apn1,
    float alpha, float delta, float r, float dr,
    int n, int half)
{
    // ---- A operand: A[m][k] = x[t0 + m*16 + 4kk + k]; lane m=n, v0/v1 = K(2*half)/(+1)
    const float* ap = xrow + t0 + n * 16 + 2 * half;
    v2f av[4];
    if (!TAIL) {
#pragma unroll
        for (int kk = 0; kk < 4; ++kk) av[kk] = *(const v2f*)(ap + 4 * kk);
    } else {
        v2f z = {0.f, 0.f};
        const bool valid = (t0 + n * 16) < T_LEN;   // whole-segment validity (T%16==0)
#pragma unroll
        for (int kk = 0; kk < 4; ++kk) av[kk] = valid ? *(const v2f*)(ap + 4 * kk) : z;
    }

    // ---- within-segment prefix: acc = X^T * W^T, K=16 as 4 chained f32 WMMAs ----
    v8f acc = {0.f, 0.f, 0.f, 0.f, 0.f, 0.f, 0.f, 0.f};
#pragma unroll
    for (int kk = 0; kk < 4; ++kk)
        acc = __builtin_amdgcn_wmma_f32_16x16x4_f32(
                  false, av[kk], false, wb[kk], (short)0, acc, false, false);

    // ---- cross-segment carry scan: c[j+1] = a^16*c[j] + e[j], e[j] = s*P[15][j] ----
    // e[j] = D[j][15]: lane 15 (j=0..7, comp j) / lane 31 (j=8..15, comp j-8); wave-uniform.
    float c = carry;
    float cv[8];
#pragma unroll
    for (int j = 0; j < 16; ++j) {
        const float ej = s * lane_read(acc[j & 7], 15 + ((j >> 3) << 4));
        if ((j >> 3) == half) cv[j & 7] = c;   // carry INTO segment j, captured by owner half
        c = q * c + ej;
    }

    // ---- finalize M and pointwise PCEN; stores coalesced per acc component ----
    // gain = (M+eps)^(-alpha)  [== exp(-a*(log eps + log1p(M/eps))), M >= 0]
#pragma unroll
    for (int v = 0; v < 8; ++v) {
        const int j  = v + 8 * half;            // segment index
        const int tp = t0 + j * 16 + n;         // position within row
        if (!TAIL || tp < T_LEN) {
            const float M    = s * acc[v] + apn1 * cv[v];
            const float xv   = xrow[tp];
            const float gain = __expf(-alpha * __logf(M + 1.0e-6f));
            const float inr  = xv * gain + delta;          // >= delta > 0
            orow[tp]         = __expf(r * __logf(inr)) - dr;
        }
    }
    return c;
}

// One wave per row (row = b*F + f). IIR smoother M_t = (1-s)M_{t-1} + s x_t as a
// blocked linear scan: 16-wide segment prefixes via V_WMMA_F32_16X16X4_F32
// (lower-triangular decay matrix W^T built in registers), readlane carry scan.
__global__ __launch_bounds__(256) void pcen_wmma_kernel(
    const float* __restrict__ x,
    const float* __restrict__ log_s,
    const float* __restrict__ log_alpha,
    const float* __restrict__ log_delta,
    const float* __restrict__ log_r,
    float* __restrict__ out,
    int rows, int bands)
{
    const int lane = threadIdx.x & 31;
    const int wave = threadIdx.x >> 5;
    const int row  = blockIdx.x * (blockDim.x >> 5) + wave;
    if (row >= rows) return;

    const int n    = lane & 15;   // column (position-in-segment) this lane owns in D
    const int half = lane >> 4;   // 0: rows 0-7 of D, 1: rows 8-15

    // ---- per-band parameters (wave-uniform) ----
    const int   f     = row % bands;
    const float s     = __expf(log_s[f]);
    const float a     = 1.0f - s;
    const float alpha = __expf(log_alpha[f]);
    const float delta = __expf(log_delta[f]);
    const float r     = __expf(log_r[f]);
    const float dr    = __expf(r * __logf(delta));    // delta^r
    const float q     = apow_int(a, 16);              // segment decay a^16
    const float apn1  = apow_int(a, n + 1);           // a^(i+1), lane position i=n

    // ---- B operand: W^T in registers. B[k][n] = W[n][4kk+k] = a^(n-4kk-k), lower-tri.
    // Assumed 4x16 f32 layout mirrors documented A layout: v0 = K(2*half), v1 = K(2*half+1).
    v2f wb[4];
#pragma unroll
    for (int kk = 0; kk < 4; ++kk) {
        const int k0 = 4 * kk + 2 * half;
        wb[kk].x = apow_int(a, n - k0);
        wb[kk].y = apow_int(a, n - k0 - 1);
    }

    const float* xrow = x   + (size_t)row * T_LEN;
    float*       orow = out + (size_t)row * T_LEN;

    // Seed: a + s == 1  =>  M_0 = a*x_0 + s*x_0 = x_0 (init_smoother_from_data)
    float carry = xrow[0];

    for (int tile = 0; tile < NT - 1; ++tile) {
        // gfx1250 global_prefetch_b8: next tile, one cacheline per lane
        __builtin_prefetch(xrow + (tile + 1) * TILE + lane * 8, 0, 1);
        carry = pcen_tile<false>(xrow, orow, tile * TILE, carry, wb,
                                 s, q, apn1, alpha, delta, r, dr, n, half);
    }
    pcen_tile<true>(xrow, orow, (NT - 1) * TILE, carry, wb,
                    s, q, apn1, alpha, delta, r, dr, n, half);
}

extern "C" void kernel_launch(void* const* d_in, const int* in_sizes, int n_in,
                              void* d_out, int out_size, void* d_ws, size_t ws_size,
                              hipStream_t stream) {
    const float* x         = (const float*)d_in[0];
    const float* log_s     = (const float*)d_in[1];
    const float* log_alpha = (const float*)d_in[2];
    const float* log_delta = (const float*)d_in[3];
    const float* log_r     = (const float*)d_in[4];
    float* out             = (float*)d_out;

    const int rows  = in_sizes[0] / T_LEN;   // B*C*F = 4096
    const int bands = in_sizes[1];           // 128

    const int threads = 256;                 // 8 wave32 per block
    const int wpb     = threads / 32;
    const int blocks  = (rows + wpb - 1) / wpb;
    pcen_wmma_kernel<<<blocks, threads, 0, stream>>>(
        x, log_s, log_alpha, log_delta, log_r, out, rows, bands);
}